// Predictor_14688788152817
// MI455X (gfx1250) — compile-verified
//
#include <hip/hip_runtime.h>
#include <hip/hip_bf16.h>

// ---------------------------------------------------------------------------
// GCN (DGL GraphConv norm='both') on gfx1250.
// Memory-bound: ~1.3 GB of gather/scatter traffic dominates; dense transforms
// use V_WMMA_F32_16X16X4_F32 (exact f32, matches reference numerics).
// ---------------------------------------------------------------------------

typedef __attribute__((ext_vector_type(2))) float v2f;
typedef __attribute__((ext_vector_type(8))) float v8f;

#define DIM 64          // IN_DIM == HID == 64
#define NGRAPH 8
#define NCLASS 2

// -------------------- generic zero fill --------------------
__global__ void k_zero(float* __restrict__ p, long long n) {
  long long i = (long long)blockIdx.x * blockDim.x + threadIdx.x;
  if (i < n) p[i] = 0.0f;
}

// -------------------- degree accumulation --------------------
__global__ void k_degrees(const int* __restrict__ esrc, const int* __restrict__ edst,
                          float* __restrict__ outdeg, float* __restrict__ indeg,
                          int n_edges) {
  int e = blockIdx.x * blockDim.x + threadIdx.x;
  if (e >= n_edges) return;
  atomicAdd(&outdeg[esrc[e]], 1.0f);
  atomicAdd(&indeg[edst[e]], 1.0f);
}

// -------------------- deg -> rsqrt(clip(deg,1)) in place --------------------
__global__ void k_norms(float* __restrict__ sn, float* __restrict__ dn, int n) {
  int i = blockIdx.x * blockDim.x + threadIdx.x;
  if (i >= n) return;
  sn[i] = rsqrtf(fmaxf(sn[i], 1.0f));
  dn[i] = rsqrtf(fmaxf(dn[i], 1.0f));
}

// -------------------- edge gather + segment-sum scatter --------------------
// one thread per (edge, dim); 64 consecutive threads share an edge so the
// h[src*64+dim] reads and agg[dst*64+dim] atomics are lane-coalesced.
__global__ void k_aggregate(const float* __restrict__ h,
                            const float* __restrict__ src_norm,
                            const int* __restrict__ esrc,
                            const int* __restrict__ edst,
                            float* __restrict__ agg,
                            long long n_work) {
  long long idx = (long long)blockIdx.x * blockDim.x + threadIdx.x;
  if (idx >= n_work) return;
  int e = (int)(idx >> 6);
  int d = (int)(idx & 63);
  int s = esrc[e];
  int t = edst[e];
  float v = h[(size_t)s * DIM + d] * src_norm[s];
  atomicAdd(&agg[(size_t)t * DIM + d], v);
}

// -------------------- WMMA node transform: relu((agg*dst_norm) @ W + b) ----
// One wave handles a 16-node tile. A (16x64, f32) lives in 16 v2f fragments;
// B streamed from W (16 KB, L2-resident); 64 V_WMMA_F32_16X16X4_F32 per wave.
__global__ void k_gemm_relu(const float* __restrict__ agg,
                            const float* __restrict__ dst_norm,
                            const float* __restrict__ W,   // [64][64] row-major
                            const float* __restrict__ b,   // [64]
                            float* __restrict__ out,       // [N][64]
                            int n_nodes) {
  const int lane = threadIdx.x & 31;
  const int wave = threadIdx.x >> 5;
  const int wavesPerBlock = blockDim.x >> 5;
  const int tile = blockIdx.x * wavesPerBlock + wave;
  const int nTiles = (n_nodes + 15) >> 4;
  if (tile >= nTiles) return;                 // wave-uniform: EXEC stays full
  const int nodeBase = tile * 16;

  const int row = lane & 15;
  const int hi  = lane >> 4;                  // selects K pair within fragment
  int nodeLd = nodeBase + row;
  if (nodeLd > n_nodes - 1) nodeLd = n_nodes - 1;   // clamp (N%16==0 anyway)
  const float dn = dst_norm[nodeLd];
  const float* arow = agg + (size_t)nodeLd * DIM;

  // A fragments: K = 0..63 in steps of 4; lane holds K = 4k + 2*hi, +1
  v2f a[16];
#pragma unroll
  for (int k = 0; k < 16; ++k) {
    int kk = k * 4 + hi * 2;
    a[k].x = arow[kk] * dn;
    a[k].y = arow[kk + 1] * dn;
  }

  const int col = lane & 15;
#pragma unroll
  for (int nt = 0; nt < 4; ++nt) {
    int n0 = nt * 16 + col;
    float bias = b[n0];                       // bias depends only on column
    v8f acc;
#pragma unroll
    for (int i = 0; i < 8; ++i) acc[i] = bias;
#pragma unroll
    for (int k = 0; k < 16; ++k) {
      int kk = k * 4 + hi * 2;
      v2f bb;
      bb.x = W[(size_t)kk * DIM + n0];
      bb.y = W[(size_t)(kk + 1) * DIM + n0];
      acc = __builtin_amdgcn_wmma_f32_16x16x4_f32(
          /*neg_a=*/false, a[k], /*neg_b=*/false, bb,
          /*c_mod=*/(short)0, acc, /*reuse_a=*/false, /*reuse_b=*/false);
    }
    // D layout: acc[i] -> row (i + 8*hi), col n0
#pragma unroll
    for (int i = 0; i < 8; ++i) {
      int m = nodeBase + i + 8 * hi;
      if (m < n_nodes) {
        float v = acc[i];
        out[(size_t)m * DIM + n0] = v > 0.0f ? v : 0.0f;
      }
    }
  }
}

// -------------------- per-graph sum pooling --------------------
__global__ void k_pool(const float* __restrict__ h, const int* __restrict__ gid,
                       float* __restrict__ hsum, float* __restrict__ cnt,
                       long long n_work) {
  long long idx = (long long)blockIdx.x * blockDim.x + threadIdx.x;
  if (idx >= n_work) return;
  int node = (int)(idx >> 6);
  int d = (int)(idx & 63);
  int g = gid[node];
  atomicAdd(&hsum[(size_t)g * DIM + d], h[(size_t)node * DIM + d]);
  if (d == 0) atomicAdd(&cnt[g], 1.0f);
}

// -------------------- head: (hsum/cnt) @ Wp + bp --------------------
__global__ void k_head(const float* __restrict__ hsum, const float* __restrict__ cnt,
                       const float* __restrict__ Wp,   // [64][2]
                       const float* __restrict__ bp,   // [2]
                       float* __restrict__ out) {      // [8][2]
  int t = threadIdx.x;
  if (t >= NGRAPH * NCLASS) return;
  int g = t >> 1;
  int c = t & 1;
  float inv = 1.0f / fmaxf(cnt[g], 1.0f);
  float s = bp[c];
#pragma unroll
  for (int d = 0; d < DIM; ++d)
    s += hsum[(size_t)g * DIM + d] * inv * Wp[(size_t)d * NCLASS + c];
  out[(size_t)g * NCLASS + c] = s;
}

// ---------------------------------------------------------------------------
extern "C" void kernel_launch(void* const* d_in, const int* in_sizes, int n_in,
                              void* d_out, int out_size, void* d_ws, size_t ws_size,
                              hipStream_t stream) {
  const float* features = (const float*)d_in[0];   // [N,64]
  const float* W1       = (const float*)d_in[1];   // [64,64]
  const float* b1       = (const float*)d_in[2];   // [64]
  const float* W2       = (const float*)d_in[3];   // [64,64]
  const float* b2       = (const float*)d_in[4];   // [64]
  const float* Wp       = (const float*)d_in[5];   // [64,2]
  const float* bp       = (const float*)d_in[6];   // [2]
  const int*   esrc     = (const int*)d_in[7];     // [E]
  const int*   edst     = (const int*)d_in[8];     // [E]
  const int*   gid      = (const int*)d_in[9];     // [N]

  const int N = in_sizes[0] / DIM;
  const int E = in_sizes[7];

  // workspace layout (floats)
  float* ws   = (float*)d_ws;
  float* srcn = ws;                         // N   (degree -> norm, in place)
  float* dstn = srcn + N;                   // N
  float* agg  = dstn + N;                   // N*64
  float* h    = agg + (size_t)N * DIM;      // N*64
  float* hsum = h + (size_t)N * DIM;        // 8*64
  float* cnt  = hsum + NGRAPH * DIM;        // 8

  const int B = 256;
  const long long zero1 = (long long)2 * N + (long long)N * DIM; // norms + agg
  const long long nwE   = (long long)E * DIM;
  const long long nwN   = (long long)N * DIM;
  const int tiles = (N + 15) / 16;
  const int gemmBlocks = (tiles + 7) / 8;   // 8 waves (16-node tiles) / block

  // degrees -> norms
  k_zero<<<(int)((zero1 + B - 1) / B), B, 0, stream>>>(srcn, zero1);
  k_degrees<<<(E + B - 1) / B, B, 0, stream>>>(esrc, edst, srcn, dstn, E);
  k_norms<<<(N + B - 1) / B, B, 0, stream>>>(srcn, dstn, N);

  // layer 1
  k_aggregate<<<(int)((nwE + B - 1) / B), B, 0, stream>>>(features, srcn, esrc, edst, agg, nwE);
  k_gemm_relu<<<gemmBlocks, B, 0, stream>>>(agg, dstn, W1, b1, h, N);

  // layer 2
  k_zero<<<(int)((nwN + B - 1) / B), B, 0, stream>>>(agg, nwN);
  k_aggregate<<<(int)((nwE + B - 1) / B), B, 0, stream>>>(h, srcn, esrc, edst, agg, nwE);
  k_gemm_relu<<<gemmBlocks, B, 0, stream>>>(agg, dstn, W2, b2, h, N);

  // mean pooling + head
  k_zero<<<1, B, 0, stream>>>(hsum, NGRAPH * DIM + NGRAPH);
  k_pool<<<(int)((nwN + B - 1) / B), B, 0, stream>>>(h, gid, hsum, cnt, nwN);
  k_head<<<1, 32, 0, stream>>>(hsum, cnt, Wp, bp, (float*)d_out);
}